// DDIOCF_44074954391993
// MI455X (gfx1250) — compile-verified
//
#include <hip/hip_runtime.h>

typedef __attribute__((ext_vector_type(2))) float v2f;
typedef __attribute__((ext_vector_type(8))) float v8f;

#define DIM 64

// acc = emb ; bufA = 0   (float4-vectorized over N*DIM/4)
__global__ void ddiocf_init(const float4* __restrict__ emb,
                            float4* __restrict__ acc,
                            float4* __restrict__ bufA, int n4) {
    int i = blockIdx.x * blockDim.x + threadIdx.x;
    if (i < n4) {
        acc[i]  = emb[i];
        bufA[i] = make_float4(0.f, 0.f, 0.f, 0.f);
    }
}

// acc += y ; nextbuf = 0
__global__ void ddiocf_accum_zero(float4* __restrict__ acc,
                                  const float4* __restrict__ y,
                                  float4* __restrict__ nextbuf, int n4) {
    int i = blockIdx.x * blockDim.x + threadIdx.x;
    if (i < n4) {
        float4 a = acc[i];
        float4 b = y[i];
        a.x += b.x; a.y += b.y; a.z += b.z; a.w += b.w;
        acc[i] = a;
        nextbuf[i] = make_float4(0.f, 0.f, 0.f, 0.f);
    }
}

// y[row[e]] += vals[e] * x[col[e]] — grid-stride wave loop, one wave32 per edge
// iteration (lane handles a float2 => 32*2 = 64 dims). Edge scalars are read
// through a readfirstlane'd uniform index so they become SMEM (s_load) reads.
// Scatter uses hardware global_atomic_add_f32 (SADDR + 32-bit voffset form)
// via inline asm so it can never expand to a CAS loop.
__global__ void ddiocf_spmm(const float* __restrict__ x,
                            const float* __restrict__ vals,
                            const int* __restrict__ rows,
                            const int* __restrict__ cols,
                            float* __restrict__ y, int nEdges) {
    int lane = threadIdx.x & 31;
    int wave = (int)((blockIdx.x * blockDim.x + threadIdx.x) >> 5);
    wave = __builtin_amdgcn_readfirstlane(wave);            // uniform wave id
    int nWaves = (int)((gridDim.x * blockDim.x) >> 5);

    for (int e = wave; e < nEdges; e += nWaves) {           // uniform scalar loop
        float v = vals[e];                                  // s_load (uniform e)
        int   r = rows[e];
        int   c = cols[e];

        float2 xv = *((const float2*)(x + (size_t)c * DIM) + lane); // 256B row gather
        float gx = v * xv.x;
        float gy = v * xv.y;

        unsigned off = ((unsigned)r * DIM + (unsigned)(lane * 2)) * 4u; // byte offset < 26MB
        asm volatile(
            "global_atomic_add_f32 %0, %1, %2\n\t"
            "global_atomic_add_f32 %0, %3, %2 offset:4"
            :
            : "v"(off), "v"(gx), "s"(y), "v"(gy)
            : "memory");
    }
}

// gamma[b] = (1/25) * ||acc[drugs[b]]||^2 via diag(X @ X^T) with WMMA f32 16x16x4.
// One wave per 16 drugs; A(16x4) and B(4x16) operands of the same tile are
// register-identical, so wmma(a, a, c) accumulates X·X^T; extract the diagonal.
__global__ void ddiocf_gamma(const float* __restrict__ acc,
                             const int* __restrict__ drugs,
                             float* __restrict__ out, int nDrugs) {
    int wave = (blockIdx.x * blockDim.x + threadIdx.x) >> 5;
    int lane = threadIdx.x & 31;
    int base = wave * 16;
    if (base >= nDrugs) return;          // wave-uniform: EXEC stays all-ones past here

    int m     = lane & 15;
    int khalf = lane >> 4;               // 0: K={0,1}+4k, 1: K={2,3}+4k
    int mi = base + m;
    if (mi >= nDrugs) mi = nDrugs - 1;   // clamp (nDrugs % 16 == 0 in practice)
    const float* row = acc + (size_t)drugs[mi] * DIM;

    v8f c = {};
    #pragma unroll
    for (int kk = 0; kk < 16; ++kk) {
        v2f a;
        a.x = row[kk * 4 + khalf * 2 + 0];
        a.y = row[kk * 4 + khalf * 2 + 1];
        // 8 args: (neg_a, A, neg_b, B, c_mod, C, reuse_a, reuse_b)
        c = __builtin_amdgcn_wmma_f32_16x16x4_f32(false, a, false, a,
                                                  (short)0, c, false, false);
    }

    const float s = 0.04f;               // (1/5)^2
    #pragma unroll
    for (int j = 0; j < 8; ++j) {
        float dj = c[j];
        if (lane == j      && (base + j)     < nDrugs) out[base + j]     = dj * s;  // D[j][j]
        if (lane == j + 24 && (base + 8 + j) < nDrugs) out[base + 8 + j] = dj * s;  // D[j+8][j+8]
    }
}

extern "C" void kernel_launch(void* const* d_in, const int* in_sizes, int n_in,
                              void* d_out, int out_size, void* d_ws, size_t ws_size,
                              hipStream_t stream) {
    const float* emb   = (const float*)d_in[0];
    const float* vals  = (const float*)d_in[1];
    const int*   rows  = (const int*)d_in[2];
    const int*   cols  = (const int*)d_in[3];
    const int*   drugs = (const int*)d_in[4];
    float* out = (float*)d_out;

    const int nEmb = in_sizes[0];        // N * 64
    const int E    = in_sizes[1];
    const int B    = in_sizes[4];
    const int n4   = nEmb / 4;

    float* acc  = (float*)d_ws;          // 3 node buffers in workspace
    float* bufA = acc  + nEmb;
    float* bufB = bufA + nEmb;

    const int eb = 256;
    const int gInit = (n4 + eb - 1) / eb;

    // Grid-stride spmm: cap at 8192 blocks (65536 waves, ~49 edges each).
    long long waveBlocksNeeded = ((long long)E * 32 + eb - 1) / eb;
    const int gSpmm = (int)(waveBlocksNeeded < 8192 ? waveBlocksNeeded : 8192);

    ddiocf_init<<<gInit, eb, 0, stream>>>((const float4*)emb, (float4*)acc, (float4*)bufA, n4);

    // d1 = G e0
    ddiocf_spmm<<<gSpmm, eb, 0, stream>>>(emb,  vals, rows, cols, bufA, E);
    ddiocf_accum_zero<<<gInit, eb, 0, stream>>>((float4*)acc, (const float4*)bufA, (float4*)bufB, n4);
    // d2 = G d1
    ddiocf_spmm<<<gSpmm, eb, 0, stream>>>(bufA, vals, rows, cols, bufB, E);
    ddiocf_accum_zero<<<gInit, eb, 0, stream>>>((float4*)acc, (const float4*)bufB, (float4*)bufA, n4);
    // d3 = G d2
    ddiocf_spmm<<<gSpmm, eb, 0, stream>>>(bufB, vals, rows, cols, bufA, E);
    ddiocf_accum_zero<<<gInit, eb, 0, stream>>>((float4*)acc, (const float4*)bufA, (float4*)bufB, n4);
    // d4 = G d3
    ddiocf_spmm<<<gSpmm, eb, 0, stream>>>(bufA, vals, rows, cols, bufB, E);
    ddiocf_accum_zero<<<gInit, eb, 0, stream>>>((float4*)acc, (const float4*)bufB, (float4*)bufA, n4);

    // gamma = diag((acc/5)(acc/5)^T) at drug rows
    const int gammaThreads = ((B + 15) / 16) * 32;           // one wave per 16 drugs
    const int gb = 128;
    ddiocf_gamma<<<(gammaThreads + gb - 1) / gb, gb, 0, stream>>>(acc, drugs, out, B);
}